// SSD_34840774705722
// MI455X (gfx1250) — compile-verified
//
#include <hip/hip_runtime.h>
#include <float.h>

#define B_ 256
#define N_ 8732
#define C_ 21
#define K_ 200
#define NPT 35                 // ceil(8732 / 256) candidates per thread
#define CONF_T 0.01f
#define NMS_T 0.45f

#if defined(__has_builtin)
# if __has_builtin(__builtin_amdgcn_global_load_async_to_lds_b32)
#  define HAVE_ASYNC_LDS 1
# endif
# if __has_builtin(__builtin_amdgcn_s_wait_asynccnt)
#  define HAVE_WAIT_ASYNC 1
# endif
#endif

typedef __attribute__((address_space(1))) int gas_int;
typedef __attribute__((address_space(3))) int las_int;

// monotonic float -> uint key (order-preserving), and its inverse
__device__ __forceinline__ unsigned int fkey(float f) {
    unsigned int u = __float_as_uint(f);
    return (u & 0x80000000u) ? ~u : (u | 0x80000000u);
}
__device__ __forceinline__ float fkey_inv(unsigned int k) {
    unsigned int u = (k & 0x80000000u) ? (k & 0x7fffffffu) : ~k;
    return __uint_as_float(u);
}

// Pass 1: per-(b,n) softmax log-sum-exp + box decode.
__global__ __launch_bounds__(256)
void ssd_prep(const float* __restrict__ loc,
              const float* __restrict__ conf,
              const float* __restrict__ dbox,
              float* __restrict__ lse,
              float* __restrict__ boxes) {
    int i = blockIdx.x * blockDim.x + threadIdx.x;
    if (i >= B_ * N_) return;
    int n = i - (i / N_) * N_;

    const float* cp = conf + (size_t)i * C_;
    float m = cp[0];
#pragma unroll
    for (int j = 1; j < C_; ++j) m = fmaxf(m, cp[j]);
    float s = 0.f;
#pragma unroll
    for (int j = 0; j < C_; ++j) s += expf(cp[j] - m);
    lse[i] = m + logf(s);

    const float4 l = *(const float4*)(loc + (size_t)i * 4);
    const float4 d = *(const float4*)(dbox + (size_t)n * 4);
    float cx = d.x + l.x * 0.1f * d.z;
    float cy = d.y + l.y * 0.1f * d.w;
    float w  = d.z * expf(l.z * 0.2f);
    float h  = d.w * expf(l.w * 0.2f);
    float4 o;
    o.x = cx - w * 0.5f;      // xymin = cxcy - wh/2
    o.y = cy - h * 0.5f;
    o.z = o.x + w;            // xymax = xymin + wh (exact reference formula)
    o.w = o.y + h;
    *(float4*)(boxes + (size_t)i * 4) = o;
}

// Pass 2: one block per (b,c): async-stage to LDS, register-resident top-k,
// NMS, compaction.
__global__ __launch_bounds__(256)
void ssd_nms(const float* __restrict__ conf,
             const float* __restrict__ lse,
             const float* __restrict__ boxes,
             float* __restrict__ out) {
    __shared__ float s_scores[N_];             // async staging buffer
    __shared__ unsigned long long s_red[8];
    __shared__ unsigned long long s_win;
    __shared__ float s_val[K_];
    __shared__ int   s_idx[K_];
    __shared__ float s_x1[K_], s_y1[K_], s_x2[K_], s_y2[K_], s_ar[K_];
    __shared__ int   s_sup[K_], s_keep[K_], s_dst[K_];
    __shared__ int   s_cnt;

    const int c   = blockIdx.x;
    const int b   = blockIdx.y;
    const int tid = threadIdx.x;
    float* outp = out + (size_t)(b * C_ + c) * (K_ * 5);

    if (c == 0) {   // per_class.at[:, 0].set(0.0)
        for (int j = tid; j < K_ * 5; j += 256) outp[j] = 0.f;
        return;
    }

    const float* base = conf + (size_t)b * N_ * C_ + c;   // stride-21 column
    const float* lrow = lse + (size_t)b * N_;

    // Stage raw conf column into LDS (CDNA5 async global->LDS DMA).
#ifdef HAVE_ASYNC_LDS
    for (int n = tid; n < N_; n += 256) {
        __builtin_amdgcn_global_load_async_to_lds_b32(
            (gas_int*)(base + (size_t)n * C_),
            (las_int*)&s_scores[n], 0, 0);
    }
# ifdef HAVE_WAIT_ASYNC
    __builtin_amdgcn_s_wait_asynccnt(0);
# else
    asm volatile("s_wait_asynccnt 0" ::: "memory");
# endif
#else
    for (int n = tid; n < N_; n += 256) s_scores[n] = base[(size_t)n * C_];
#endif
    __syncthreads();

    // Build per-thread candidate keys in registers:
    // key = fkey(masked_score) << 32 | (N - idx); 0 = empty/consumed slot.
    unsigned long long key[NPT];
    unsigned long long myBest = 0ull;
#pragma unroll
    for (int j = 0; j < NPT; ++j) {
        int idx = tid + j * 256;
        unsigned long long kk = 0ull;
        if (idx < N_) {
            float sc = expf(s_scores[idx] - lrow[idx]);
            float masked = (sc > CONF_T) ? sc : -1.0f;
            kk = ((unsigned long long)fkey(masked) << 32) |
                 (unsigned int)(N_ - idx);          // smaller idx wins ties
        }
        key[j] = kk;
        myBest = (kk > myBest) ? kk : myBest;
    }

    // ---- top-K: 200 tournament extractions ----
    for (int k = 0; k < K_; ++k) {
        unsigned long long wb = myBest;
#pragma unroll
        for (int off = 16; off > 0; off >>= 1) {
            unsigned long long o = __shfl_xor(wb, off, 32);
            wb = (o > wb) ? o : wb;
        }
        if ((tid & 31) == 0) s_red[tid >> 5] = wb;
        __syncthreads();
        if (tid == 0) {
            unsigned long long bb = s_red[0];
#pragma unroll
            for (int wv = 1; wv < 8; ++wv)
                if (s_red[wv] > bb) bb = s_red[wv];
            s_win = bb;
            int idx = N_ - (int)(unsigned int)(bb & 0xFFFFFFFFull);
            s_idx[k] = idx;
            s_val[k] = fkey_inv((unsigned int)(bb >> 32));
        }
        __syncthreads();
        const unsigned long long bb = s_win;
        const int widx = N_ - (int)(unsigned int)(bb & 0xFFFFFFFFull);
        if (tid == (widx & 255)) {          // winning thread: consume its slot
            const int slot = widx >> 8;
            unsigned long long nb = 0ull;
#pragma unroll
            for (int j = 0; j < NPT; ++j) {
                if (j == slot) key[j] = 0ull;
                nb = (key[j] > nb) ? key[j] : nb;
            }
            myBest = nb;
        }
    }

    // ---- gather boxes for the 200 candidates ----
    if (tid < K_) {
        int idx = s_idx[tid];
        const float4 bx = *(const float4*)(boxes + ((size_t)b * N_ + idx) * 4);
        s_x1[tid] = bx.x; s_y1[tid] = bx.y; s_x2[tid] = bx.z; s_y2[tid] = bx.w;
        s_ar[tid] = (bx.z - bx.x) * (bx.w - bx.y);
        s_sup[tid]  = (s_val[tid] > CONF_T) ? 0 : 1;   // suppressed0 = !valid
        s_keep[tid] = 0;
    }

    // ---- sequential NMS, inner loop parallel over candidates ----
    for (int i = 0; i < K_; ++i) {
        __syncthreads();
        int active = s_sup[i] ? 0 : 1;
        if (tid == 0) s_keep[i] = active;
        if (active && tid < K_ && tid != i) {
            float xx1 = fmaxf(s_x1[tid], s_x1[i]);
            float yy1 = fmaxf(s_y1[tid], s_y1[i]);
            float xx2 = fmaxf(s_x2[tid], s_x2[i]);   // reference uses max here too
            float yy2 = fmaxf(s_y2[tid], s_y2[i]);
            float inter = (xx2 - xx1) * (yy2 - yy1); // unclamped, per reference
            float uni   = s_ar[tid] - inter + s_ar[i];
            float iou   = inter / uni;
            if (iou > NMS_T) s_sup[tid] = 1;
        }
    }
    __syncthreads();

    // ---- compaction (cumsum(keep)-1) ----
    if (tid == 0) {
        int cnt = 0;
        for (int k2 = 0; k2 < K_; ++k2)
            s_dst[k2] = s_keep[k2] ? cnt++ : -1;
        s_cnt = cnt;
    }
    __syncthreads();

    const int cnt = s_cnt;
    for (int j = tid; j < K_; j += 256) {
        if (j >= cnt) {
            float* r = outp + (size_t)j * 5;
            r[0] = 0.f; r[1] = 0.f; r[2] = 0.f; r[3] = 0.f; r[4] = 0.f;
        }
    }
    if (tid < K_ && s_keep[tid]) {
        float* r = outp + (size_t)s_dst[tid] * 5;
        r[0] = s_val[tid];
        r[1] = s_x1[tid]; r[2] = s_y1[tid];
        r[3] = s_x2[tid]; r[4] = s_y2[tid];
    }
}

extern "C" void kernel_launch(void* const* d_in, const int* in_sizes, int n_in,
                              void* d_out, int out_size, void* d_ws, size_t ws_size,
                              hipStream_t stream) {
    (void)in_sizes; (void)n_in; (void)out_size; (void)ws_size;
    const float* loc  = (const float*)d_in[0];   // (256, 8732, 4)
    const float* conf = (const float*)d_in[1];   // (256, 8732, 21)
    const float* dbox = (const float*)d_in[2];   // (8732, 4)
    float* out = (float*)d_out;                  // (256, 21, 200, 5)

    float* lse   = (float*)d_ws;                 // B*N floats   (8.9 MB)
    float* boxes = lse + (size_t)B_ * N_;        // B*N*4 floats (35.8 MB)

    const int total = B_ * N_;
    ssd_prep<<<(total + 255) / 256, 256, 0, stream>>>(loc, conf, dbox, lse, boxes);

    dim3 grid(C_, B_);   // classes fastest -> same-batch blocks share L2 lines
    ssd_nms<<<grid, 256, 0, stream>>>(conf, lse, boxes, out);
}